// LP_83468394431056
// MI455X (gfx1250) — compile-verified
//
#include <hip/hip_runtime.h>
#include <stdint.h>

// Label propagation: out = clip(mask ? y : ALPHA*segment_sum(out[src]*w, dst) + (1-ALPHA)*out, 0, 1), K times.
// N=100000, C=64, E=3200000, K=10. Working set ~115MB -> fits MI455X 192MB L2.
// L2/atomic-bound; wave32 layout: 1 wave per edge, float2 per lane (32*2 = 64 channels).

#define C_CHANNELS 64
#define K_ITERS 10
#define ALPHA_F 0.9f
#define TILE_EDGES 256
#define BLK 256

#if defined(__HIP_DEVICE_COMPILE__) && \
    __has_builtin(__builtin_amdgcn_global_load_async_to_lds_b32) && \
    __has_builtin(__builtin_amdgcn_s_wait_asynccnt)
#define USE_ASYNC_LDS 1
#else
#define USE_ASYNC_LDS 0
#endif

#define AS1 __attribute__((address_space(1)))
#define AS3 __attribute__((address_space(3)))

__global__ __launch_bounds__(BLK) void lp_init_out(
    const float* __restrict__ y, const int* __restrict__ mask,
    float* __restrict__ out, int total4)
{
  int i = blockIdx.x * BLK + threadIdx.x;
  if (i >= total4) return;
  int node = i / (C_CHANNELS / 4);
  float4 v = make_float4(0.f, 0.f, 0.f, 0.f);
  if (mask[node] != 0) v = reinterpret_cast<const float4*>(y)[i];
  reinterpret_cast<float4*>(out)[i] = v;
}

__global__ __launch_bounds__(BLK) void lp_zero(float* __restrict__ agg, int total4)
{
  int i = blockIdx.x * BLK + threadIdx.x;
  if (i < total4) reinterpret_cast<float4*>(agg)[i] = make_float4(0.f, 0.f, 0.f, 0.f);
}

__global__ __launch_bounds__(BLK) void lp_update(
    const float* __restrict__ y, const int* __restrict__ mask,
    const float* __restrict__ agg, float* __restrict__ out, int total4)
{
  int i = blockIdx.x * BLK + threadIdx.x;
  if (i >= total4) return;
  int node = i / (C_CHANNELS / 4);
  float4 r;
  if (mask[node] != 0) {
    r = reinterpret_cast<const float4*>(y)[i];
  } else {
    float4 a = reinterpret_cast<const float4*>(agg)[i];
    float4 o = reinterpret_cast<const float4*>(out)[i];
    r.x = fminf(fmaxf(ALPHA_F * a.x + (1.0f - ALPHA_F) * o.x, 0.f), 1.f);
    r.y = fminf(fmaxf(ALPHA_F * a.y + (1.0f - ALPHA_F) * o.y, 0.f), 1.f);
    r.z = fminf(fmaxf(ALPHA_F * a.z + (1.0f - ALPHA_F) * o.z, 0.f), 1.f);
    r.w = fminf(fmaxf(ALPHA_F * a.w + (1.0f - ALPHA_F) * o.w, 0.f), 1.f);
  }
  reinterpret_cast<float4*>(out)[i] = r;
}

// One wave per edge; lane handles channels [2*lane, 2*lane+1].
// Edge records double-buffered in LDS via gfx1250 async global->LDS copies.
__global__ __launch_bounds__(BLK) void lp_edge_scatter(
    const int* __restrict__ src, const int* __restrict__ dst,
    const float* __restrict__ ew, const float* __restrict__ out,
    float* __restrict__ agg, int E)
{
  __shared__ uint4 rec[2][TILE_EDGES];   // interleaved {src, dst, w, pad}, 8KB
  const int tid  = threadIdx.x;
  const int lane = tid & 31;
  const int wave = tid >> 5;
  const int numTiles = (E + TILE_EDGES - 1) / TILE_EDGES;

  auto stage = [&](int tile, int buf) {
    int e = tile * TILE_EDGES + tid;
    if (e >= E) e = E - 1;   // clamp: duplicate record, masked out at processing
#if USE_ASYNC_LDS
    AS3 int* l = (AS3 int*)&rec[buf][tid];
    __builtin_amdgcn_global_load_async_to_lds_b32(
        (AS1 int*)(src + e), l + 0, 0, 0);
    __builtin_amdgcn_global_load_async_to_lds_b32(
        (AS1 int*)(dst + e), l + 1, 0, 0);
    __builtin_amdgcn_global_load_async_to_lds_b32(
        (AS1 int*)(ew + e),  l + 2, 0, 0);
#else
    uint32_t* l = (uint32_t*)&rec[buf][tid];
    l[0] = (uint32_t)src[e];
    l[1] = (uint32_t)dst[e];
    l[2] = __float_as_uint(ew[e]);
#endif
  };

  int tile = blockIdx.x;
  int buf = 0;
  if (tile < numTiles) stage(tile, 0);

  for (; tile < numTiles; tile += gridDim.x) {
    int nxt = tile + gridDim.x;
    if (nxt < numTiles) {
      stage(nxt, buf ^ 1);           // prefetch next tile into other buffer
#if USE_ASYNC_LDS
      __builtin_amdgcn_s_wait_asynccnt(3);   // current tile's 3 copies complete
#endif
    } else {
#if USE_ASYNC_LDS
      __builtin_amdgcn_s_wait_asynccnt(0);
#endif
    }
#if USE_ASYNC_LDS
    asm volatile("" ::: "memory");
#endif

    // Each wave processes the 32 edges it staged itself -> no cross-wave barrier needed.
    const int base = tile * TILE_EDGES + wave * 32;
    int cnt = E - base;
    if (cnt > 32) cnt = 32;
#pragma unroll 4
    for (int j = 0; j < cnt; ++j) {
      uint4 r = rec[buf][wave * 32 + j];          // uniform addr -> ds_load_b128 broadcast
      int s = (int)r.x;
      int d = (int)r.y;
      float wt = __uint_as_float(r.z);
      const float2 v = *reinterpret_cast<const float2*>(
          out + (size_t)s * C_CHANNELS + lane * 2);        // coalesced 256B row gather
      float* ap = agg + (size_t)d * C_CHANNELS + lane * 2; // coalesced 256B row scatter
      unsafeAtomicAdd(ap + 0, v.x * wt);   // native global_atomic_add_f32
      unsafeAtomicAdd(ap + 1, v.y * wt);
    }
    buf ^= 1;
  }
}

extern "C" void kernel_launch(void* const* d_in, const int* in_sizes, int n_in,
                              void* d_out, int out_size, void* d_ws, size_t ws_size,
                              hipStream_t stream) {
  const float* y    = (const float*)d_in[0];
  const int*   mask = (const int*)  d_in[1];
  const int*   ei   = (const int*)  d_in[2];
  const float* ew   = (const float*)d_in[3];

  const int E = in_sizes[3];          // edge_weight count
  const int N = in_sizes[1];          // train_mask count
  const int* srcp = ei;               // edge_index row 0
  const int* dstp = ei + E;           // edge_index row 1

  float* out = (float*)d_out;         // out lives in d_out (N*C f32)
  float* agg = (float*)d_ws;          // scratch: N*C f32 = 25.6 MB

  const int total4 = (N * C_CHANNELS) / 4;
  const int gElem = (total4 + BLK - 1) / BLK;
  const int numTiles = (E + TILE_EDGES - 1) / TILE_EDGES;
  int gEdge = (numTiles + 3) / 4;     // ~4 tiles per block -> pipelined double-buffer
  if (gEdge < 1) gEdge = 1;

  lp_init_out<<<gElem, BLK, 0, stream>>>(y, mask, out, total4);
  for (int k = 0; k < K_ITERS; ++k) {
    lp_zero<<<gElem, BLK, 0, stream>>>(agg, total4);
    lp_edge_scatter<<<gEdge, BLK, 0, stream>>>(srcp, dstp, ew, out, agg, E);
    lp_update<<<gElem, BLK, 0, stream>>>(y, mask, agg, out, total4);
  }
}